// Prob2_22282290332146
// MI455X (gfx1250) — compile-verified
//
#include <hip/hip_runtime.h>
#include <hip/hip_bf16.h>

typedef __attribute__((ext_vector_type(2))) float v2f;
typedef __attribute__((ext_vector_type(8))) float v8f;

#define BN_EPS 1e-5f

// conn[o*6+i] = position j such that CUSTOM[i][j] == o, else -1
__constant__ int d_conn[96] = {
   0, 0, 0,-1,-1,-1,
  -1, 1, 1, 0,-1,-1,
  -1,-1, 2, 1, 0,-1,
  -1,-1,-1, 2, 1, 0,
   1,-1,-1,-1, 2, 1,
   2, 2,-1,-1,-1, 2,
   3, 3, 3, 3,-1,-1,
  -1, 4, 4, 4, 3,-1,
  -1,-1, 5, 5, 4, 3,
   4,-1,-1, 6, 5, 4,
   5, 5,-1,-1, 6, 5,
   6, 6, 6,-1,-1, 6,
   7, 7,-1, 7, 7,-1,
  -1, 8, 7,-1, 8, 7,
   8,-1, 8, 8,-1, 8,
   9, 9, 9, 9, 9, 9
};
__constant__ float d_counts[16] = {3,3,3,3,3,3,4,4,4,4,4,4,4,4,4,6};

// ---- workspace layout (float offsets) ----
#define OFF_BN1_SUM 0      // 6
#define OFF_BN1_SQ  8      // 6
#define OFF_BN1_SC  16     // 6
#define OFF_BN1_SH  24     // 6
#define OFF_BN2_SUM 32     // 16
#define OFF_BN2_SQ  48     // 16
#define OFF_BN2_SC  64     // 16
#define OFF_BN2_SH  80     // 16
#define OFF_BN3_SUM 96     // 128
#define OFF_BN3_SQ  224    // 128
#define OFF_BN3_SC  352    // 128
#define OFF_BN3_SH  480    // 128
#define OFF_WC3     512    // 2400  (dense 16x6x5x5)
#define OFF_WC5T    2912   // 400*128 transposed, padded
#define OFF_F6WT    54112  // 120*96 transposed, padded
#define OFF_H2      65632  // B*6*14*14
#define OFF_H4      9699424UL   // B*400
#define OFF_H5      12976224UL  // B*120
#define OFF_H6      13959264UL  // B*84

// ---------------- prep: build dense/transposed weight tensors ----------------
__global__ void k_prep(const float* __restrict__ c3w, const float* __restrict__ c5w,
                       const float* __restrict__ f6w,
                       float* __restrict__ wc3, float* __restrict__ wc5t,
                       float* __restrict__ f6wt) {
  int idx = blockIdx.x * blockDim.x + threadIdx.x;
  if (idx < 2400) {                       // wc3[o][i][25]
    int o = idx / 150, rem = idx % 150, i = rem / 25, k = rem % 25;
    int j = d_conn[o * 6 + i];
    wc3[idx] = (j >= 0) ? c3w[j * 150 + i * 25 + k] : 0.f;
  }
  if (idx < 51200) {                      // wc5t[k][n], n padded to 128
    int k = idx / 128, n = idx % 128;
    wc5t[idx] = (n < 120) ? c5w[n * 400 + k] : 0.f;
  }
  if (idx < 11520) {                      // f6wt[k][n], n padded to 96
    int k = idx / 96, n = idx % 96;
    f6wt[idx] = (n < 84) ? f6w[n * 120 + k] : 0.f;
  }
}

// ---------------- C1 pass 1: conv+bias, accumulate BN1 stats ----------------
__global__ void k_c1_stats(const float* __restrict__ x, const float* __restrict__ w,
                           const float* __restrict__ bias,
                           float* __restrict__ sum, float* __restrict__ sq, int total) {
  __shared__ float sred[12];
  if (threadIdx.x < 12) sred[threadIdx.x] = 0.f;
  __syncthreads();
  float ls[6] = {0,0,0,0,0,0}, lq[6] = {0,0,0,0,0,0};
  for (int idx = blockIdx.x * blockDim.x + threadIdx.x; idx < total;
       idx += gridDim.x * blockDim.x) {
    int b = idx / 784, p = idx % 784;
    int oy = p / 28, ox = p % 28;
    const float* xb = x + b * 1024 + oy * 32 + ox;
    float patch[25];
#pragma unroll
    for (int ky = 0; ky < 5; ++ky)
#pragma unroll
      for (int kx = 0; kx < 5; ++kx) patch[ky * 5 + kx] = xb[ky * 32 + kx];
#pragma unroll
    for (int c = 0; c < 6; ++c) {
      float acc = bias[c];
#pragma unroll
      for (int k = 0; k < 25; ++k) acc += patch[k] * w[c * 25 + k];
      ls[c] += acc; lq[c] += acc * acc;
    }
  }
#pragma unroll
  for (int c = 0; c < 6; ++c) { atomicAdd(&sred[c], ls[c]); atomicAdd(&sred[6 + c], lq[c]); }
  __syncthreads();
  if (threadIdx.x < 6) {
    atomicAdd(&sum[threadIdx.x], sred[threadIdx.x]);
    atomicAdd(&sq[threadIdx.x], sred[6 + threadIdx.x]);
  }
}

// ---------------- BN finalize: scale/shift from sums ----------------
__global__ void k_bn_final(const float* __restrict__ sum, const float* __restrict__ sq,
                           const float* __restrict__ g, const float* __restrict__ bb,
                           float* __restrict__ sc, float* __restrict__ sh,
                           int C, float invN) {
  int c = blockIdx.x * blockDim.x + threadIdx.x;
  if (c < C) {
    float m = sum[c] * invN;
    float v = sq[c] * invN - m * m;
    float s = g[c] * rsqrtf(v + BN_EPS);
    sc[c] = s;
    sh[c] = bb[c] - m * s;
  }
}

// ------- C1 pass 2: conv+bias -> BN -> ReLU -> maxpool2 -> S2 affine -> ReLU -------
__global__ void k_c1_apply(const float* __restrict__ x, const float* __restrict__ w,
                           const float* __restrict__ bias,
                           const float* __restrict__ sc, const float* __restrict__ sh,
                           const float* __restrict__ s2w, const float* __restrict__ s2b,
                           float* __restrict__ h2, int total) {
  int idx = blockIdx.x * blockDim.x + threadIdx.x;
  if (idx >= total) return;
  int b = idx / 196, p = idx % 196;
  int py = p / 14, px = p % 14;
  const float* xb = x + b * 1024 + (2 * py) * 32 + 2 * px;
  float patch[36];
#pragma unroll
  for (int yy = 0; yy < 6; ++yy)
#pragma unroll
    for (int xx = 0; xx < 6; ++xx) patch[yy * 6 + xx] = xb[yy * 32 + xx];
#pragma unroll
  for (int c = 0; c < 6; ++c) {
    float mx = -1e30f;
#pragma unroll
    for (int dy = 0; dy < 2; ++dy)
#pragma unroll
      for (int dx = 0; dx < 2; ++dx) {
        float acc = bias[c];
#pragma unroll
        for (int ky = 0; ky < 5; ++ky)
#pragma unroll
          for (int kx = 0; kx < 5; ++kx)
            acc += patch[(dy + ky) * 6 + (dx + kx)] * w[c * 25 + ky * 5 + kx];
        float v = fmaxf(0.f, sc[c] * acc + sh[c]);
        mx = fmaxf(mx, v);
      }
    h2[(b * 6 + c) * 196 + p] = fmaxf(0.f, mx * s2w[c] + s2b[c]);
  }
}

// ---------------- C3 pass 1: dense conv, accumulate BN2 stats ----------------
__global__ void k_c3_stats(const float* __restrict__ h2, const float* __restrict__ wc3,
                           const float* __restrict__ c3b,
                           float* __restrict__ sum, float* __restrict__ sq, int total) {
  __shared__ float swc3[2400];
  __shared__ float sred[32];
  for (int i = threadIdx.x; i < 2400; i += blockDim.x) swc3[i] = wc3[i];
  if (threadIdx.x < 32) sred[threadIdx.x] = 0.f;
  __syncthreads();
  float ls[16], lq[16];
#pragma unroll
  for (int c = 0; c < 16; ++c) { ls[c] = 0.f; lq[c] = 0.f; }
  for (int idx = blockIdx.x * blockDim.x + threadIdx.x; idx < total;
       idx += gridDim.x * blockDim.x) {
    int b = idx / 100, p = idx % 100;
    int oy = p / 10, ox = p % 10;
    float acc[16];
#pragma unroll
    for (int c = 0; c < 16; ++c) acc[c] = c3b[c] * d_counts[c];
    for (int i = 0; i < 6; ++i) {
      const float* hp = h2 + (b * 6 + i) * 196 + oy * 14 + ox;
      float patch[25];
#pragma unroll
      for (int ky = 0; ky < 5; ++ky)
#pragma unroll
        for (int kx = 0; kx < 5; ++kx) patch[ky * 5 + kx] = hp[ky * 14 + kx];
#pragma unroll
      for (int c = 0; c < 16; ++c) {
        const float* wp = &swc3[(c * 6 + i) * 25];
        float d = 0.f;
#pragma unroll
        for (int k = 0; k < 25; ++k) d += patch[k] * wp[k];
        acc[c] += d;
      }
    }
#pragma unroll
    for (int c = 0; c < 16; ++c) { ls[c] += acc[c]; lq[c] += acc[c] * acc[c]; }
  }
#pragma unroll
  for (int c = 0; c < 16; ++c) { atomicAdd(&sred[c], ls[c]); atomicAdd(&sred[16 + c], lq[c]); }
  __syncthreads();
  if (threadIdx.x < 16) {
    atomicAdd(&sum[threadIdx.x], sred[threadIdx.x]);
    atomicAdd(&sq[threadIdx.x], sred[16 + threadIdx.x]);
  }
}

// ------- C3 pass 2: conv -> BN2 -> ReLU -> maxpool2 -> S4 affine -> ReLU -------
__global__ void __launch_bounds__(256)
k_c3_apply(const float* __restrict__ h2, const float* __restrict__ wc3,
           const float* __restrict__ c3b,
           const float* __restrict__ sc, const float* __restrict__ sh,
           const float* __restrict__ s4w, const float* __restrict__ s4b,
           float* __restrict__ h4, int total) {
  __shared__ float swc3[2400];
  for (int i = threadIdx.x; i < 2400; i += blockDim.x) swc3[i] = wc3[i];
  __syncthreads();
  int idx = blockIdx.x * blockDim.x + threadIdx.x;
  if (idx >= total) return;
  int b = idx / 25, p = idx % 25;
  int py = p / 5, px = p % 5;
  float acc[64];
#pragma unroll
  for (int c = 0; c < 16; ++c) {
    float init = c3b[c] * d_counts[c];
#pragma unroll
    for (int q = 0; q < 4; ++q) acc[c * 4 + q] = init;
  }
  for (int i = 0; i < 6; ++i) {
    const float* hp = h2 + (b * 6 + i) * 196 + (2 * py) * 14 + 2 * px;
    float patch[36];
#pragma unroll
    for (int yy = 0; yy < 6; ++yy)
#pragma unroll
      for (int xx = 0; xx < 6; ++xx) patch[yy * 6 + xx] = hp[yy * 14 + xx];
#pragma unroll
    for (int c = 0; c < 16; ++c) {
      const float* wp = &swc3[(c * 6 + i) * 25];
#pragma unroll
      for (int dy = 0; dy < 2; ++dy)
#pragma unroll
        for (int dx = 0; dx < 2; ++dx) {
          float d = 0.f;
#pragma unroll
          for (int ky = 0; ky < 5; ++ky)
#pragma unroll
            for (int kx = 0; kx < 5; ++kx)
              d += patch[(dy + ky) * 6 + (dx + kx)] * wp[ky * 5 + kx];
          acc[c * 4 + dy * 2 + dx] += d;
        }
    }
  }
#pragma unroll
  for (int c = 0; c < 16; ++c) {
    float mx = -1e30f;
#pragma unroll
    for (int q = 0; q < 4; ++q)
      mx = fmaxf(mx, fmaxf(0.f, sc[c] * acc[c * 4 + q] + sh[c]));
    h4[b * 400 + c * 25 + p] = fmaxf(0.f, mx * s4w[c] + s4b[c]);
  }
}

// ---------------- C5: [B,400] x [400,120] GEMM with V_WMMA_F32_16X16X4_F32 ----------------
// Fused: + bias, store h5raw, accumulate BN3 per-channel stats.
__global__ void k_c5_wmma(const float* __restrict__ h4, const float* __restrict__ wc5t,
                          const float* __restrict__ c5b, float* __restrict__ h5,
                          float* __restrict__ sum, float* __restrict__ sq) {
  int wave = (blockIdx.x * blockDim.x + threadIdx.x) >> 5;
  int lane = threadIdx.x & 31;
  int mt = wave >> 3, nt = wave & 7;          // 512 M-tiles x 8 N-tiles (120 pad 128)
  int m0 = mt * 16, n0 = nt * 16;
  int row = lane & 15;
  int kk = (lane >> 4) << 1;                  // lanes 0-15 -> K 0,1 ; lanes 16-31 -> K 2,3
  const float* aptr = h4 + (m0 + row) * 400 + kk;
  const float* bptr = wc5t + kk * 128 + n0 + row;
  v8f c = {};
  for (int k = 0; k < 400; k += 4) {
    v2f a, b;
    a.x = aptr[k];       a.y = aptr[k + 1];
    b.x = bptr[k * 128]; b.y = bptr[(k + 1) * 128];
    c = __builtin_amdgcn_wmma_f32_16x16x4_f32(false, a, false, b, (short)0, c, false, false);
  }
  int n = n0 + row;
  bool valid = (n < 120);
  float bias = valid ? c5b[n] : 0.f;
  int mrow = m0 + ((lane >> 4) << 3);
  float lsum = 0.f, lsq = 0.f;
#pragma unroll
  for (int r = 0; r < 8; ++r) {
    float v = c[r] + bias;
    if (valid) {
      h5[(mrow + r) * 120 + n] = v;
      lsum += v; lsq += v * v;
    }
  }
  if (valid) { atomicAdd(&sum[n], lsum); atomicAdd(&sq[n], lsq); }
}

// ---------------- F6: BN3+ReLU folded into A-load; [B,120] x [120,84] WMMA; +bias, ReLU ----------------
__global__ void k_f6_wmma(const float* __restrict__ h5,
                          const float* __restrict__ sc3, const float* __restrict__ sh3,
                          const float* __restrict__ f6wt, const float* __restrict__ f6b,
                          float* __restrict__ h6) {
  int wave = (blockIdx.x * blockDim.x + threadIdx.x) >> 5;
  int lane = threadIdx.x & 31;
  int mt = wave / 6, nt = wave % 6;           // 512 M-tiles x 6 N-tiles (84 pad 96)
  int m0 = mt * 16, n0 = nt * 16;
  int row = lane & 15;
  int kk = (lane >> 4) << 1;
  const float* aptr = h5 + (m0 + row) * 120 + kk;
  const float* bptr = f6wt + kk * 96 + n0 + row;
  v8f c = {};
  for (int k = 0; k < 120; k += 4) {
    int ch0 = k + kk, ch1 = k + kk + 1;
    v2f a, b;
    a.x = fmaxf(0.f, sc3[ch0] * aptr[k]     + sh3[ch0]);
    a.y = fmaxf(0.f, sc3[ch1] * aptr[k + 1] + sh3[ch1]);
    b.x = bptr[k * 96];
    b.y = bptr[(k + 1) * 96];
    c = __builtin_amdgcn_wmma_f32_16x16x4_f32(false, a, false, b, (short)0, c, false, false);
  }
  int n = n0 + row;
  if (n < 84) {
    float bias = f6b[n];
    int mrow = m0 + ((lane >> 4) << 3);
#pragma unroll
    for (int r = 0; r < 8; ++r)
      h6[(mrow + r) * 84 + n] = fmaxf(0.f, c[r] + bias);
  }
}

// ---------------- RBF: squared distance to 10 prototypes ----------------
__global__ void k_rbf(const float* __restrict__ h6, const float* __restrict__ proto,
                      float* __restrict__ out, int total) {
  int idx = blockIdx.x * blockDim.x + threadIdx.x;
  if (idx >= total) return;
  int b = idx / 10, j = idx % 10;
  const float* h = h6 + b * 84;
  const float* p = proto + j * 84;
  float d = 0.f;
#pragma unroll 4
  for (int i = 0; i < 84; ++i) { float t = h[i] - p[i]; d += t * t; }
  out[b * 10 + j] = d;
}

extern "C" void kernel_launch(void* const* d_in, const int* in_sizes, int n_in,
                              void* d_out, int out_size, void* d_ws, size_t ws_size,
                              hipStream_t stream) {
  const float* x     = (const float*)d_in[0];
  const float* c1_w  = (const float*)d_in[1];
  const float* c1_b  = (const float*)d_in[2];
  const float* bn1_g = (const float*)d_in[3];
  const float* bn1_b = (const float*)d_in[4];
  const float* s2_w  = (const float*)d_in[5];
  const float* s2_b  = (const float*)d_in[6];
  const float* c3_w  = (const float*)d_in[7];
  const float* c3_b  = (const float*)d_in[8];
  const float* bn2_g = (const float*)d_in[9];
  const float* bn2_b = (const float*)d_in[10];
  const float* s4_w  = (const float*)d_in[11];
  const float* s4_b  = (const float*)d_in[12];
  const float* c5_w  = (const float*)d_in[13];
  const float* c5_b  = (const float*)d_in[14];
  const float* bn3_g = (const float*)d_in[15];
  const float* bn3_b = (const float*)d_in[16];
  const float* f6_w  = (const float*)d_in[17];
  const float* f6_b  = (const float*)d_in[18];
  const float* rbf_p = (const float*)d_in[19];
  float* ws  = (float*)d_ws;
  float* out = (float*)d_out;

  const int B = in_sizes[0] / 1024;   // 8192

  // zero BN accumulators (graph-capturable)
  hipMemsetAsync(d_ws, 0, 512 * sizeof(float), stream);

  // prep transposed/dense weights
  k_prep<<<(51200 + 255) / 256, 256, 0, stream>>>(
      c3_w, c5_w, f6_w, ws + OFF_WC3, ws + OFF_WC5T, ws + OFF_F6WT);

  // C1 stats -> BN1 -> apply (+pool +S2)
  k_c1_stats<<<2048, 256, 0, stream>>>(x, c1_w, c1_b,
      ws + OFF_BN1_SUM, ws + OFF_BN1_SQ, B * 784);
  k_bn_final<<<1, 64, 0, stream>>>(ws + OFF_BN1_SUM, ws + OFF_BN1_SQ, bn1_g, bn1_b,
      ws + OFF_BN1_SC, ws + OFF_BN1_SH, 6, 1.f / (float)(B * 784));
  k_c1_apply<<<(B * 196 + 255) / 256, 256, 0, stream>>>(x, c1_w, c1_b,
      ws + OFF_BN1_SC, ws + OFF_BN1_SH, s2_w, s2_b, ws + OFF_H2, B * 196);

  // C3 stats -> BN2 -> apply (+pool +S4)
  k_c3_stats<<<1600, 256, 0, stream>>>(ws + OFF_H2, ws + OFF_WC3, c3_b,
      ws + OFF_BN2_SUM, ws + OFF_BN2_SQ, B * 100);
  k_bn_final<<<1, 64, 0, stream>>>(ws + OFF_BN2_SUM, ws + OFF_BN2_SQ, bn2_g, bn2_b,
      ws + OFF_BN2_SC, ws + OFF_BN2_SH, 16, 1.f / (float)(B * 100));
  k_c3_apply<<<(B * 25 + 255) / 256, 256, 0, stream>>>(ws + OFF_H2, ws + OFF_WC3, c3_b,
      ws + OFF_BN2_SC, ws + OFF_BN2_SH, s4_w, s4_b, ws + OFF_H4, B * 25);

  // C5 GEMM (WMMA f32) + bias + BN3 stats
  k_c5_wmma<<<(B / 16) * 8 / 8, 256, 0, stream>>>(ws + OFF_H4, ws + OFF_WC5T, c5_b,
      ws + OFF_H5, ws + OFF_BN3_SUM, ws + OFF_BN3_SQ);
  k_bn_final<<<1, 128, 0, stream>>>(ws + OFF_BN3_SUM, ws + OFF_BN3_SQ, bn3_g, bn3_b,
      ws + OFF_BN3_SC, ws + OFF_BN3_SH, 120, 1.f / (float)B);

  // F6 GEMM (WMMA f32, BN3+ReLU folded into A-load) + bias + ReLU
  k_f6_wmma<<<(B / 16) * 6 / 8, 256, 0, stream>>>(ws + OFF_H5,
      ws + OFF_BN3_SC, ws + OFF_BN3_SH, ws + OFF_F6WT, f6_b, ws + OFF_H6);

  // RBF output
  k_rbf<<<(B * 10 + 255) / 256, 256, 0, stream>>>(ws + OFF_H6, rbf_p, out, B * 10);
}